// PhysicsInformedLoss_790273982717
// MI455X (gfx1250) — compile-verified
//
#include <hip/hip_runtime.h>
#include <hip/hip_bf16.h>

#define WBINS 4096
#define NBLK 128
#define TPB 1024
#define EPSF 1e-6f
#define ALPHA_C 0.1f
#define BETA_C 0.1f
#define CAP_C 1000.0f

typedef __attribute__((ext_vector_type(2))) float v2f;
typedef __attribute__((ext_vector_type(8))) float v8f;

// Wave32 full-lane sum using V_WMMA_F32_16X16X4_F32.
// A (16x4 f32): lane L<16 holds A[L][0]=a.x, A[L][1]=a.y; lane L>=16 holds A[L-16][2],[3].
// With a = {x, 0} and B = ones, D[m][n] = x[lane m] + x[lane m+16] for every column n.
// C/D layout: lanes 0-15 hold rows 0..7 in VGPRs 0..7; lanes 16-31 hold rows 8..15.
// So sum of the 8 accumulator regs + shfl_xor(16) = full 32-lane sum in every lane.
// Requires EXEC == all 1s: call only from converged full blocks.
__device__ __forceinline__ float wave_sum32_wmma(float x) {
  v2f a; a[0] = x;    a[1] = 0.0f;
  v2f b; b[0] = 1.0f; b[1] = 1.0f;
  v8f c = {};
  c = __builtin_amdgcn_wmma_f32_16x16x4_f32(false, a, false, b, (short)0, c, false, false);
  float s = ((c[0] + c[1]) + (c[2] + c[3])) + ((c[4] + c[5]) + (c[6] + c[7]));
  s += __shfl_xor(s, 16, 32);
  return s;
}

__device__ __forceinline__ float block_sum(float x, float* red) {
  const int lane = threadIdx.x & 31;
  const int wv   = threadIdx.x >> 5;
  const int nw   = blockDim.x >> 5;
  float s = wave_sum32_wmma(x);
  if (lane == 0) red[wv] = s;
  __syncthreads();
  if (wv == 0) {
    float v = (lane < nw) ? red[lane] : 0.0f;
    v = wave_sum32_wmma(v);
    if (lane == 0) red[0] = v;
  }
  __syncthreads();
  float r = red[0];
  __syncthreads();
  return r;
}

__global__ void k_zero(unsigned int* p, int count) {
  int i = blockIdx.x * blockDim.x + threadIdx.x;
  if (i < count) p[i] = 0u;
}

// ---- Main fused pass: CE sums + per-window segment sums + level-1 radix histogram + key dump
__global__ void k_main(const float2* __restrict__ logits,
                       const int*   __restrict__ y,
                       const unsigned char* __restrict__ mask,
                       const float4* __restrict__ xraw,
                       const int*   __restrict__ widx,
                       const float* __restrict__ cw,
                       unsigned int* __restrict__ keys,
                       float* __restrict__ blockseg,   // [NBLK][5*WBINS] (4 f32 arrays + 1 u32 array)
                       float* __restrict__ blockscal,  // [NBLK][4]
                       int n) {
  extern __shared__ unsigned char smem[];
  float*        s_cnt = (float*)smem;
  float*        s_p   = s_cnt + WBINS;
  float*        s_rt  = s_p   + WBINS;
  float*        s_pd  = s_rt  + WBINS;
  unsigned int* s_h   = (unsigned int*)(s_pd + WBINS);
  float*        s_red = (float*)(s_h + WBINS);   // 32 floats

  for (int i = threadIdx.x; i < WBINS; i += blockDim.x) {
    s_cnt[i] = 0.f; s_p[i] = 0.f; s_rt[i] = 0.f; s_pd[i] = 0.f; s_h[i] = 0u;
  }
  __syncthreads();

  const float cw0 = cw[0], cw1 = cw[1];
  float sm = 0.f, sd = 0.f, sn = 0.f;
  const int stride = gridDim.x * blockDim.x;
  for (int i = blockIdx.x * blockDim.x + threadIdx.x; i < n; i += stride) {
    __builtin_prefetch(&xraw[i + stride], 0, 1);   // global_prefetch_b8, speculative
    float2 lg = logits[i];
    int    yy = y[i];
    float  mk = mask[i] ? 1.0f : 0.0f;
    float4 xr = xraw[i];
    int    wi = widx[i];

    // stable log-softmax on 2 classes
    float mx  = fmaxf(lg.x, lg.y);
    float lse = mx + __logf(__expf(lg.x - mx) + __expf(lg.y - mx));
    float nll = lse - (yy ? lg.y : lg.x);
    float wy  = yy ? cw1 : cw0;
    sm += mk;
    sd += mk * wy;
    sn += mk * wy * nll;

    float ap  = 1.0f / (1.0f + __expf(lg.x - lg.y));  // softmax[:,1]
    float fr  = fmaxf(xr.w, 0.0f);                    // pkt_rate
    float dob = fmaxf(xr.z, 0.0f);                    // D_observed
    bool  valid = (wi >= 0) && (mk > 0.0f);

    unsigned int key = valid ? __float_as_uint(dob) : 0xFFFFFFFFu;
    keys[i] = key;

    if (valid) {
      atomicAdd(&s_cnt[wi], 1.0f);
      atomicAdd(&s_p[wi],   ap);
      atomicAdd(&s_rt[wi],  ap * fr);
      atomicAdd(&s_pd[wi],  ap * dob);
      atomicAdd(&s_h[key >> 20], 1u);
    }
  }
  __syncthreads();

  // deterministic per-block flush (no global float atomics)
  float* dst = blockseg + (size_t)blockIdx.x * (5 * WBINS);
  for (int i = threadIdx.x; i < WBINS; i += blockDim.x) {
    dst[i]              = s_cnt[i];
    dst[WBINS + i]      = s_p[i];
    dst[2 * WBINS + i]  = s_rt[i];
    dst[3 * WBINS + i]  = s_pd[i];
    ((unsigned int*)dst)[4 * WBINS + i] = s_h[i];
  }

  // WMMA-based block scalar reduction
  float bm = block_sum(sm, s_red);
  float bd = block_sum(sd, s_red);
  float bn = block_sum(sn, s_red);
  if (threadIdx.x == 0) {
    float* bs = blockscal + blockIdx.x * 4;
    bs[0] = bm; bs[1] = bd; bs[2] = bn; bs[3] = 0.f;
  }
}

// ---- Combine per-block segment partials (deterministic)
__global__ void k_segreduce(const float* __restrict__ blockseg,
                            float* __restrict__ segs,
                            unsigned int* __restrict__ hist1) {
  int idx = blockIdx.x * blockDim.x + threadIdx.x;
  if (idx >= 5 * WBINS) return;
  if (idx < 4 * WBINS) {
    float s = 0.f;
    for (int b = 0; b < NBLK; ++b) s += blockseg[(size_t)b * 5 * WBINS + idx];
    segs[idx] = s;
  } else {
    int j = idx - 4 * WBINS;
    const unsigned int* bs = (const unsigned int*)blockseg;
    unsigned int s = 0u;
    for (int b = 0; b < NBLK; ++b) s += bs[(size_t)b * 5 * WBINS + 4 * WBINS + j];
    hist1[j] = s;
  }
}

// ---- Level-1 scan: total n, ranks, level-1 bin + residual rank for lo/hi
__global__ void k_scan1(const unsigned int* __restrict__ hist1,
                        const float* __restrict__ blockscal,
                        unsigned int* __restrict__ sel,
                        float* __restrict__ scal) {
  if (threadIdx.x != 0 || blockIdx.x != 0) return;
  double m = 0.0, d = 0.0, nm = 0.0;
  for (int b = 0; b < NBLK; ++b) {
    m  += blockscal[b * 4 + 0];
    d  += blockscal[b * 4 + 1];
    nm += blockscal[b * 4 + 2];
  }
  scal[0] = (float)m; scal[1] = (float)d; scal[2] = (float)nm;

  unsigned long long ncnt = 0;
  for (int i = 0; i < WBINS; ++i) ncnt += hist1[i];
  sel[8] = (unsigned int)ncnt;

  double pos = 0.75 * (double)((ncnt > 0) ? (ncnt - 1) : 0);
  if (pos < 0.0) pos = 0.0;
  unsigned long long lo = (unsigned long long)floor(pos);
  unsigned long long hi = (unsigned long long)ceil(pos);
  ((float*)sel)[9] = (float)(pos - (double)lo);

  unsigned long long acc = 0;
  unsigned int blo = 0, bhi = 0, rlo = 0, rhi = 0;
  bool flo = false, fhi = false;
  for (int i = 0; i < WBINS; ++i) {
    unsigned long long c = hist1[i];
    if (!flo && acc + c > lo) { blo = i; rlo = (unsigned int)(lo - acc); flo = true; }
    if (!fhi && acc + c > hi) { bhi = i; rhi = (unsigned int)(hi - acc); fhi = true; }
    acc += c;
  }
  sel[0] = blo; sel[1] = bhi; sel[2] = rlo; sel[3] = rhi;
}

// ---- Level-2 refine: histogram bits[19:8] of keys in the candidate level-1 bins
__global__ void k_pass2(const unsigned int* __restrict__ keys,
                        const unsigned int* __restrict__ sel,
                        unsigned int* __restrict__ hist2, int n) {
  unsigned int blo = sel[0], bhi = sel[1];
  const int stride = gridDim.x * blockDim.x;
  for (int i = blockIdx.x * blockDim.x + threadIdx.x; i < n; i += stride) {
    unsigned int k   = keys[i];
    unsigned int top = k >> 20;
    unsigned int mid = (k >> 8) & 0xFFFu;
    if (top == blo) atomicAdd(&hist2[mid], 1u);
    if (top == bhi) atomicAdd(&hist2[WBINS + mid], 1u);
  }
}

__global__ void k_scan2(const unsigned int* __restrict__ hist2,
                        unsigned int* __restrict__ sel) {
  if (threadIdx.x != 0 || blockIdx.x != 0) return;
  for (int j = 0; j < 2; ++j) {
    unsigned int r = sel[2 + j];
    const unsigned int* h = hist2 + j * WBINS;
    unsigned int acc = 0, b = 0, rr = 0; bool f = false;
    for (int i = 0; i < WBINS; ++i) {
      unsigned int c = h[i];
      if (!f && acc + c > r) { b = i; rr = r - acc; f = true; }
      acc += c;
    }
    sel[4 + j] = b; sel[6 + j] = rr;
  }
}

// ---- Level-3 refine: histogram bits[7:0] among keys matching top-24 prefix
__global__ void k_pass3(const unsigned int* __restrict__ keys,
                        const unsigned int* __restrict__ sel,
                        unsigned int* __restrict__ hist3, int n) {
  unsigned int plo = (sel[0] << 12) | sel[4];
  unsigned int phi = (sel[1] << 12) | sel[5];
  const int stride = gridDim.x * blockDim.x;
  for (int i = blockIdx.x * blockDim.x + threadIdx.x; i < n; i += stride) {
    unsigned int k = keys[i];
    unsigned int t = k >> 8;
    if (t == plo) atomicAdd(&hist3[k & 0xFFu], 1u);
    if (t == phi) atomicAdd(&hist3[256 + (k & 0xFFu)], 1u);
  }
}

// ---- Final: exact quantile + per-window physics terms + outputs (4 f32)
__global__ void k_final(const float* __restrict__ segs,
                        const unsigned int* __restrict__ hist3,
                        const unsigned int* __restrict__ sel,
                        const float* __restrict__ scal,
                        float* __restrict__ out) {
  __shared__ float s_red[32];
  __shared__ float sh_ref;
  if (threadIdx.x == 0) {
    unsigned int nvalid = sel[8];
    float frac = ((const float*)sel)[9];
    unsigned int klo = 0, khi = 0;
    {
      unsigned int r = sel[6], acc = 0, b = 0; bool f = false;
      for (int i = 0; i < 256; ++i) { unsigned int c = hist3[i];
        if (!f && acc + c > r) { b = i; f = true; } acc += c; }
      klo = (sel[0] << 20) | (sel[4] << 8) | b;
    }
    {
      unsigned int r = sel[7], acc = 0, b = 0; bool f = false;
      for (int i = 0; i < 256; ++i) { unsigned int c = hist3[256 + i];
        if (!f && acc + c > r) { b = i; f = true; } acc += c; }
      khi = (sel[1] << 20) | (sel[5] << 8) | b;
    }
    float vlo = __uint_as_float(klo), vhi = __uint_as_float(khi);
    float q = vlo * (1.0f - frac) + vhi * frac;
    sh_ref = (nvalid > 0) ? fmaxf(q, EPSF) : 1.0f;
  }
  __syncthreads();
  const float ref = sh_ref;

  float inc_s = 0.f, flow_s = 0.f, lat_s = 0.f;
  for (int w = threadIdx.x; w < WBINS; w += blockDim.x) {
    float cnt = segs[w];
    float sp  = segs[WBINS + w];
    float ar  = segs[2 * WBINS + w];
    float spd = segs[3 * WBINS + w];
    float incl  = (cnt >= 2.0f && sp >= EPSF) ? 1.0f : 0.0f;
    float dmean = spd / (sp + EPSF);
    float rr    = ar / (CAP_C + EPSF);
    float bu    = fmaxf(rr - 1.0f, 0.0f);
    float ft    = bu * bu;
    float rho   = fminf(fmaxf(rr, 0.0f), 0.99f);
    float dth   = 1.0f / (1.0f - rho + EPSF);
    float lt    = fmaxf(dth - dmean / ref, 0.0f);
    inc_s  += incl;
    flow_s += ft * incl;
    lat_s  += lt * incl;
  }
  float n_inc = block_sum(inc_s,  s_red);
  float fsum  = block_sum(flow_s, s_red);
  float lsum  = block_sum(lat_s,  s_red);

  if (threadIdx.x == 0) {
    float smk = scal[0], sden = scal[1], snum = scal[2];
    float l_data = snum / fmaxf(sden, 1e-12f);
    float safe   = fmaxf(n_inc, 1.0f);
    float l_flow = (n_inc > 0.0f) ? fsum / safe : 0.0f;
    float l_lat  = (n_inc > 0.0f) ? lsum / safe : 0.0f;
    float any    = (smk > 0.0f) ? 1.0f : 0.0f;
    l_data *= any; l_flow *= any; l_lat *= any;
    out[0] = l_data + ALPHA_C * l_flow + BETA_C * l_lat;
    out[1] = l_data;
    out[2] = l_flow;
    out[3] = l_lat;
  }
}

extern "C" void kernel_launch(void* const* d_in, const int* in_sizes, int n_in,
                              void* d_out, int out_size, void* d_ws, size_t ws_size,
                              hipStream_t stream) {
  (void)n_in; (void)out_size; (void)ws_size;
  const float2*        logits = (const float2*)d_in[0];
  const int*           y      = (const int*)d_in[1];
  const unsigned char* mask   = (const unsigned char*)d_in[2];
  const float4*        xraw   = (const float4*)d_in[3];
  const int*           widx   = (const int*)d_in[4];
  const float*         cw     = (const float*)d_in[5];
  float* out = (float*)d_out;
  const int n = in_sizes[1];

  unsigned char* ws = (unsigned char*)d_ws;
  float*        segs      = (float*)(ws + 0);          // 4*WBINS f32          = 64 KB
  unsigned int* hist1     = (unsigned int*)(ws + 65536);   // WBINS u32       = 16 KB
  unsigned int* hist2     = (unsigned int*)(ws + 81920);   // 2*WBINS u32     = 32 KB
  unsigned int* hist3     = (unsigned int*)(ws + 114688);  // 512 u32         =  2 KB
  float*        scal      = (float*)(ws + 116736);         // 16 f32
  unsigned int* sel       = (unsigned int*)(ws + 116800);  // 16 u32
  float*        blockscal = (float*)(ws + 116864);         // NBLK*4 f32
  unsigned int* keys      = (unsigned int*)(ws + 131072);  // n u32
  size_t boff = (131072 + (size_t)n * 4 + 255) & ~(size_t)255;
  float* blockseg = (float*)(ws + boff);                   // NBLK * 5*WBINS

  // zero the atomically-updated refine histograms (hist2 + hist3 contiguous: 8704 words)
  k_zero<<<34, 256, 0, stream>>>(hist2, 2 * WBINS + 512);

  const size_t smem_main = (size_t)(4 * WBINS) * 4 + (size_t)WBINS * 4 + 32 * 4; // 82048 B
  k_main<<<NBLK, TPB, smem_main, stream>>>(logits, y, mask, xraw, widx, cw,
                                           keys, blockseg, blockscal, n);
  k_segreduce<<<(5 * WBINS + 255) / 256, 256, 0, stream>>>(blockseg, segs, hist1);
  k_scan1<<<1, 32, 0, stream>>>(hist1, blockscal, sel, scal);
  k_pass2<<<512, 256, 0, stream>>>(keys, sel, hist2, n);
  k_scan2<<<1, 32, 0, stream>>>(hist2, sel);
  k_pass3<<<512, 256, 0, stream>>>(keys, sel, hist3, n);
  k_final<<<1, 1024, 0, stream>>>(segs, hist3, sel, scal, out);
}